// HybridNN_44727789420662
// MI455X (gfx1250) — compile-verified
//
#include <hip/hip_runtime.h>
#include <hip/hip_bf16.h>
#include <math.h>

// ---------------------------------------------------------------------------
// HybridNN: quantum-circuit-in-LDS + FP32-WMMA GEMMs for MI455X (gfx1250)
//
// Sizes: B=512, COM_Q=6, IMG_Q=11, ALL_Q=12, DEC_OUT = 11*2048 = 22528
//   K1: com = x(512x2048) @ W_cls(2048x64) + b_cls        [f32 WMMA 16x16x4]
//   K2: 512 independent 12-qubit statevector sims in LDS  -> energy[512]
//       (angle tables staged via GLOBAL_LOAD_ASYNC_TO_LDS, ASYNCcnt)
//   K3: out = tanh(relu(energy*W1+b1) @ W2 + b2)          [f32 WMMA 16x16x4,
//       2 N-tiles per wave, global_prefetch on W2 rows]
// ---------------------------------------------------------------------------

typedef float v2f __attribute__((ext_vector_type(2)));
typedef float v8f __attribute__((ext_vector_type(8)));

#define NB 512
#define NK_CLS 2048
#define NCOM 64
#define NHID 128
#define NDEC 22528          // 11 * 2048
#define NSTATE 4096         // 2^12
#define NPAIR 2048

// ---- gfx1250 async global->LDS helpers (inline asm; ASYNCcnt-tracked) -----
__device__ __forceinline__ void async_g2lds_b128(void* lds, const void* gaddr) {
  const unsigned l = (unsigned)(uintptr_t)lds;   // low 32 bits = LDS offset
  asm volatile("global_load_async_to_lds_b128 %0, %1, off"
               :: "v"(l), "v"(gaddr) : "memory");
}
__device__ __forceinline__ void wait_async0() {
  asm volatile("s_wait_asynccnt 0x0" ::: "memory");
}

// ============================ K1: com_params GEMM ==========================
// grid: 128 blocks x 32 threads; each wave computes one 16x16 tile of (512x64)
__global__ __launch_bounds__(32) void k_com_gemm(const float* __restrict__ x,
                                                 const float* __restrict__ Wc,
                                                 const float* __restrict__ bc,
                                                 float* __restrict__ com) {
  const int tile = blockIdx.x;      // 0..127
  const int tm = tile >> 2;         // 32 M-tiles
  const int tn = tile & 3;          // 4  N-tiles
  const int lane = threadIdx.x;     // 0..31
  const int mn = lane & 15;
  const int koff = (lane < 16) ? 0 : 2;

  const float* __restrict__ arow = x + (size_t)(tm * 16 + mn) * NK_CLS;
  const float* __restrict__ bcol = Wc + (tn * 16 + mn);

  v8f c = {};
  for (int k0 = 0; k0 < NK_CLS; k0 += 4) {
    v2f a, b;
    a.x = arow[k0 + koff];
    a.y = arow[k0 + koff + 1];
    b.x = bcol[(size_t)(k0 + koff) * NCOM];
    b.y = bcol[(size_t)(k0 + koff + 1) * NCOM];
    c = __builtin_amdgcn_wmma_f32_16x16x4_f32(false, a, false, b,
                                              (short)0, c, false, false);
  }
  const int n = tn * 16 + mn;
  const float bias = bc[n];
  const int mbase = tm * 16 + ((lane < 16) ? 0 : 8);
#pragma unroll
  for (int r = 0; r < 8; ++r)
    com[(size_t)(mbase + r) * NCOM + n] = c[r] + bias;
}

// ============================ K2: circuit sim ==============================
__device__ __forceinline__ void apply1q(float* re, float* im, int w,
                                        float u00r, float u00i, float u01r, float u01i,
                                        float u10r, float u10i, float u11r, float u11i,
                                        int tid) {
  const int bit = 1 << (11 - w);
  const int lowmask = bit - 1;
  for (int p = tid; p < NPAIR; p += 256) {
    const int i0 = ((p & ~lowmask) << 1) | (p & lowmask);
    const int i1 = i0 | bit;
    const float a0r = re[i0], a0i = im[i0];
    const float a1r = re[i1], a1i = im[i1];
    re[i0] = u00r * a0r - u00i * a0i + u01r * a1r - u01i * a1i;
    im[i0] = u00r * a0i + u00i * a0r + u01r * a1i + u01i * a1r;
    re[i1] = u10r * a0r - u10i * a0i + u11r * a1r - u11i * a1i;
    im[i1] = u10r * a0i + u10i * a0r + u11r * a1i + u11i * a1r;
  }
  __syncthreads();
}

__device__ __forceinline__ void applyH(float* re, float* im, int w, int tid) {
  const float s = 0.70710678118654752f;
  apply1q(re, im, w, s, 0.f, s, 0.f, s, 0.f, -s, 0.f, tid);
}

__device__ __forceinline__ void applyCnot(float* re, float* im, int c, int t, int tid) {
  const int cb = 1 << (11 - c);
  const int tb = 1 << (11 - t);
  for (int i = tid; i < NSTATE; i += 256) {
    if ((i & cb) && !(i & tb)) {
      const int j = i | tb;
      const float r0 = re[i], q0 = im[i];
      re[i] = re[j]; im[i] = im[j];
      re[j] = r0;    im[j] = q0;
    }
  }
  __syncthreads();
}

// uniformly-controlled RY on qubit 0, controls = qubits 1..q (theta in LDS)
__device__ __forceinline__ void applyUcry(float* re, float* im,
                                          const float* theta, int q, int tid) {
  const int shift = 11 - q;
  for (int p = tid; p < NPAIR; p += 256) {   // p = index with bit11 = 0
    const int i1 = p | NPAIR;
    const float th = 0.5f * theta[p >> shift];
    const float c = __cosf(th);
    const float s = __sinf(th);
    const float a0r = re[p],  a0i = im[p];
    const float a1r = re[i1], a1i = im[i1];
    re[p]  = c * a0r - s * a1r;  im[p]  = c * a0i - s * a1i;
    re[i1] = s * a0r + c * a1r;  im[i1] = s * a0i + c * a1i;
  }
  __syncthreads();
}

__global__ __launch_bounds__(256) void k_circuit(const float* __restrict__ x,
                                                 const float* __restrict__ com,
                                                 const float* __restrict__ asz,
                                                 float* __restrict__ energy) {
  __shared__ float re[NSTATE];
  __shared__ float im[NSTATE];
  __shared__ __align__(16) float thx[2048];   // x row (ucry q=11 angles)
  __shared__ __align__(16) float thc[64];     // com row (ucry q=6 angles)
  __shared__ float U[24][8];                  // rot matrices
  __shared__ float red[256];

  const int b = blockIdx.x;
  const int tid = threadIdx.x;

  // async-stage angle tables straight into LDS (no VGPR round trip)
  {
    const float* xrow = x + (size_t)b * 2048;
    for (int i = tid; i < 512; i += 256)            // 512 x b128 = 8KB
      async_g2lds_b128(&thx[i * 4], xrow + i * 4);
    if (tid < 16)                                   // 16 x b128 = 256B
      async_g2lds_b128(&thc[tid * 4], com + (size_t)b * 64 + tid * 4);
  }

  // init |0...0> while async copies are in flight
  for (int i = tid; i < NSTATE; i += 256) { re[i] = 0.f; im[i] = 0.f; }
  if (tid == 0) re[0] = 1.0f;
  // per-block rotation matrices from asz (2x12x3)
  if (tid < 24) {
    const float phi = asz[tid * 3 + 0];
    const float th  = asz[tid * 3 + 1];
    const float om  = asz[tid * 3 + 2];
    const float c = __cosf(0.5f * th), s = __sinf(0.5f * th);
    const float apo = 0.5f * (phi + om), amo = 0.5f * (phi - om);
    const float ear = __cosf(apo), eai = -__sinf(apo);
    const float ebr = __cosf(amo), ebi = -__sinf(amo);
    U[tid][0] =  ear * c;  U[tid][1] =  eai * c;   // U00 = ea*c
    U[tid][2] = -ebr * s;  U[tid][3] =  ebi * s;   // U01 = -conj(eb)*s
    U[tid][4] =  ebr * s;  U[tid][5] =  ebi * s;   // U10 = eb*s
    U[tid][6] =  ear * c;  U[tid][7] = -eai * c;   // U11 = conj(ea)*c
  }
  wait_async0();      // per-wave: all async LDS writes landed
  __syncthreads();    // workgroup-visible

  // H on qubits 0..COM_Q
  for (int w = 0; w <= 6; ++w) applyH(re, im, w, tid);
  // uc-RY with com_params (controls = qubits 1..6)
  applyUcry(re, im, thc, 6, tid);
  // entangling layers
  for (int l = 0; l < 2; ++l) {
    for (int w = 0; w < 12; ++w) {
      const float* u = U[l * 12 + w];
      apply1q(re, im, w, u[0], u[1], u[2], u[3], u[4], u[5], u[6], u[7], tid);
    }
    const int r = (l % 11) + 1;   // l=0 -> 1, l=1 -> 2
    for (int i = 0; i < 12; ++i) applyCnot(re, im, i, (i + r) % 12, tid);
  }
  // H on qubits 1..IMG_Q
  for (int w = 1; w <= 11; ++w) applyH(re, im, w, tid);
  // uc-RY with img params (controls = qubits 1..11)
  applyUcry(re, im, thx, 11, tid);

  // energy = -<Z0> = sum_{bit11=1}|a|^2 - sum_{bit11=0}|a|^2
  float acc = 0.f;
  for (int i = tid; i < NSTATE; i += 256) {
    const float v = re[i] * re[i] + im[i] * im[i];
    acc += (i & NPAIR) ? v : -v;
  }
  red[tid] = acc;
  __syncthreads();
  for (int s = 128; s > 0; s >>= 1) {
    if (tid < s) red[tid] += red[tid + s];
    __syncthreads();
  }
  if (tid == 0) energy[b] = red[0];
}

// ============================ K3: decoder GEMM =============================
// grid: (88, 32) x 256 threads. Block = 16 rows x 256 cols of output.
// 8 waves/block, each wave a 16x32 slab (two 16x16 tiles sharing one A frag).
__global__ __launch_bounds__(256) void k_decoder(const float* __restrict__ energy,
                                                 const float* __restrict__ W1,
                                                 const float* __restrict__ b1,
                                                 const float* __restrict__ W2,
                                                 const float* __restrict__ b2,
                                                 float* __restrict__ out) {
  __shared__ float hA[16 * NHID];   // h[m][k] for this block's 16 batch rows

  const int mt = blockIdx.y;        // 0..31
  const int tid = threadIdx.x;

  // h = relu(energy * W1 + b1), staged once per block
  for (int idx = tid; idx < 16 * NHID; idx += 256) {
    const int m = idx >> 7;
    const int k = idx & 127;
    const float hv = energy[mt * 16 + m] * W1[k] + b1[k];
    hA[idx] = hv > 0.f ? hv : 0.f;
  }
  __syncthreads();

  const int wave = tid >> 5;                       // 0..7
  const int lane = tid & 31;
  const int ntp = (blockIdx.x * 8 + wave) * 2;     // first of 2 N-tiles
  const int n0 = ntp * 16 + (lane & 15);
  const int n1 = n0 + 16;
  const int koff = (lane < 16) ? 0 : 2;
  const int m = lane & 15;

  const float* __restrict__ w2c0 = W2 + n0;
  const float* __restrict__ w2c1 = W2 + n1;
  v8f c0 = {}, c1 = {};
#pragma unroll 4
  for (int k0 = 0; k0 < NHID; k0 += 4) {
    if (k0 + 4 < NHID) {   // pull next B rows toward the WGP (L2 -> near)
      __builtin_prefetch(w2c0 + (size_t)(k0 + 4 + koff) * NDEC, 0, 3);
    }
    v2f a, b0, b1v;
    a.x = hA[m * NHID + k0 + koff];
    a.y = hA[m * NHID + k0 + koff + 1];
    b0.x  = w2c0[(size_t)(k0 + koff) * NDEC];
    b0.y  = w2c0[(size_t)(k0 + koff + 1) * NDEC];
    b1v.x = w2c1[(size_t)(k0 + koff) * NDEC];
    b1v.y = w2c1[(size_t)(k0 + koff + 1) * NDEC];
    c0 = __builtin_amdgcn_wmma_f32_16x16x4_f32(false, a, false, b0,
                                               (short)0, c0, false, false);
    c1 = __builtin_amdgcn_wmma_f32_16x16x4_f32(false, a, false, b1v,
                                               (short)0, c1, false, false);
  }

  const float bias0 = b2[n0];
  const float bias1 = b2[n1];
  const int mbase = mt * 16 + ((lane < 16) ? 0 : 8);
#pragma unroll
  for (int r = 0; r < 8; ++r) {
    {
      const float v = c0[r] + bias0;
      const float e = __expf(2.f * fabsf(v));
      out[(size_t)(mbase + r) * NDEC + n0] = copysignf(1.f - 2.f / (e + 1.f), v);
    }
    {
      const float v = c1[r] + bias1;
      const float e = __expf(2.f * fabsf(v));
      out[(size_t)(mbase + r) * NDEC + n1] = copysignf(1.f - 2.f / (e + 1.f), v);
    }
  }
}

// ============================ launcher =====================================
extern "C" void kernel_launch(void* const* d_in, const int* in_sizes, int n_in,
                              void* d_out, int out_size, void* d_ws, size_t ws_size,
                              hipStream_t stream) {
  const float* x   = (const float*)d_in[0];
  const float* Wc  = (const float*)d_in[1];
  const float* bc  = (const float*)d_in[2];
  const float* asz = (const float*)d_in[3];
  const float* W1  = (const float*)d_in[4];
  const float* b1  = (const float*)d_in[5];
  const float* W2  = (const float*)d_in[6];
  const float* b2  = (const float*)d_in[7];
  float* out = (float*)d_out;

  float* com    = (float*)d_ws;            // 512*64 f32
  float* energy = com + NB * NCOM;         // 512 f32

  k_com_gemm<<<128, 32, 0, stream>>>(x, Wc, bc, com);
  k_circuit<<<NB, 256, 0, stream>>>(x, com, asz, energy);
  k_decoder<<<dim3(NDEC / (16 * 16), NB / 16), 256, 0, stream>>>(energy, W1, b1, W2, b2, out);
}